// LightweightConv1dTBC_60138132078903
// MI455X (gfx1250) — compile-verified
//
#include <hip/hip_runtime.h>

typedef float v2f __attribute__((ext_vector_type(2)));
typedef float v8f __attribute__((ext_vector_type(8)));

#define T_DIM 2048
#define B_DIM 8
#define C_DIM 1024
#define H_DIM 16
#define K_TAPS 31
#define G_DIM (C_DIM / H_DIM)   // 64 channels per head

#define SEGS 16                 // T split into 16 segments of 128
#define TILES_PER_SEG 8         // 8 tiles of 16 t-rows each
#define CBLOCKS (C_DIM / 16)    // 64
#define WAVES_PER_BLOCK 8
#define TOTAL_WAVES (SEGS * B_DIM * CBLOCKS)            // 8192
#define GRID_BLOCKS (TOTAL_WAVES / WAVES_PER_BLOCK)     // 1024

#define ROWS (B_DIM * C_DIM)    // 8192 floats per t-row (32 KB)

// ---------------------------------------------------------------------------
// Kernel 1: per-head softmax of the (H,1,K) weight -> d_ws (H*K floats)
// ---------------------------------------------------------------------------
__global__ void lwconv_softmax_w(const float* __restrict__ w,
                                 float* __restrict__ sw) {
    int h = threadIdx.x;
    if (h >= H_DIM) return;
    float v[K_TAPS];
    float m = -1e30f;
    #pragma unroll
    for (int k = 0; k < K_TAPS; ++k) {
        v[k] = w[h * K_TAPS + k];
        m = fmaxf(m, v[k]);
    }
    float s = 0.0f;
    #pragma unroll
    for (int k = 0; k < K_TAPS; ++k) {
        v[k] = __expf(v[k] - m);
        s += v[k];
    }
    float inv = 1.0f / s;
    #pragma unroll
    for (int k = 0; k < K_TAPS; ++k) sw[h * K_TAPS + k] = v[k] * inv;
}

// ---------------------------------------------------------------------------
// Kernel 2: Toeplitz-WMMA depthwise causal conv.
// Window layout: out[i] = sum_{j'=0..47} Wt'[i][j'] * x[t0-32+j'],
// Wt'[i][j'] = w[j'-2-i] (zero-padded columns at the FRONT: j'=0,1), so the
// per-tile new chunks (j'=32..47) read exactly rows t0..t0+15 -> never OOB.
// One base pointer per wave; all hot-loop offsets are compile-time immediates.
// ---------------------------------------------------------------------------
__global__ __launch_bounds__(256) void lwconv_wmma_kernel(
    const float* __restrict__ x,     // [T,B,C]
    const float* __restrict__ sw,    // [H,K] softmaxed weights
    const float* __restrict__ bias,  // [C]
    float* __restrict__ out)         // [T,B,C]
{
    const int lane = threadIdx.x & 31;
    const int wave = blockIdx.x * WAVES_PER_BLOCK + (threadIdx.x >> 5);

    const int cb  = wave & (CBLOCKS - 1);        // c-block
    const int b   = (wave >> 6) & (B_DIM - 1);   // batch
    const int seg = wave >> 9;                   // T segment

    const int n     = lane & 15;                 // matrix column / A row
    const int half  = (lane >> 4) << 1;          // 0 (lanes 0-15) or 2 (16-31)
    const int cbase = cb * 16;
    const int head  = cbase / G_DIM;
    const int ch    = cbase + n;
    const int t0    = seg * (TILES_PER_SEG * 16);

    // --- 12 constant A chunks: A[q][m][kk] = w[(4q+kk) - 2 - m] ------------
    const float* swh = sw + head * K_TAPS;
    v2f A[12];
    #pragma unroll
    for (int q = 0; q < 12; ++q) {
        int tap0 = 4 * q + half - 2 - n;
        int tap1 = tap0 + 1;
        int c0 = min(max(tap0, 0), K_TAPS - 1);
        int c1 = min(max(tap1, 0), K_TAPS - 1);
        float a0 = swh[c0];
        float a1 = swh[c1];
        A[q].x = (tap0 == c0) ? a0 : 0.0f;
        A[q].y = (tap1 == c1) ? a1 : 0.0f;
    }

    const float bs = bias[ch];

    const long long baseOff = (long long)b * C_DIM + ch;
    // Window base: row t0-32 (+ per-lane half), single pointer for the wave.
    const float* xseg = x + baseOff + (long long)(t0 - 32 + half) * ROWS;
    // Output base: row t0 (+ per-lane half-of-C/D rows), single pointer.
    float* oseg = out + baseOff + (long long)(t0 + ((lane >> 4) << 3)) * ROWS;

    // --- Preamble: chunks 0..7 cover rows t0-32 .. t0-1 --------------------
    v2f Bw[12];
    if (seg != 0) {
        #pragma unroll
        for (int q = 0; q < 8; ++q) {
            Bw[q].x = xseg[(4 * q) * ROWS];
            Bw[q].y = xseg[(4 * q + 1) * ROWS];
        }
    } else {
        // Only segment 0 can touch negative rows: clamp + select (one-time).
        const float* xb = x + baseOff;
        #pragma unroll
        for (int q = 0; q < 8; ++q) {
            int r0 = -32 + 4 * q + half;
            int r1 = r0 + 1;
            float v0 = xb[(long long)max(r0, 0) * ROWS];
            float v1 = xb[(long long)max(r1, 0) * ROWS];
            Bw[q].x = (r0 >= 0) ? v0 : 0.0f;
            Bw[q].y = (r1 >= 0) ? v1 : 0.0f;
        }
    }

    #pragma unroll
    for (int tile = 0; tile < TILES_PER_SEG; ++tile) {
        // New chunks 8..11: rows t0 + 16*tile .. t0 + 16*tile + 15 (in-bounds)
        const int tb = 32 + 16 * tile;   // row offset from window base
        #pragma unroll
        for (int dq = 0; dq < 4; ++dq) {
            Bw[8 + dq].x = xseg[(tb + 4 * dq) * ROWS];
            Bw[8 + dq].y = xseg[(tb + 4 * dq + 1) * ROWS];
        }

        v8f acc = { bs, bs, bs, bs, bs, bs, bs, bs };
        #pragma unroll
        for (int q = 0; q < 12; ++q) {
            acc = __builtin_amdgcn_wmma_f32_16x16x4_f32(
                false, A[q], false, Bw[q], (short)0, acc, false, false);
        }

        // Store 16x16 tile: vgpr r -> row 16*tile + r (immediate offsets)
        #pragma unroll
        for (int r = 0; r < 8; ++r) {
            oseg[(16 * tile + r) * ROWS] = acc[r];
        }

        // Slide window: advance by 4 chunks (16 t-rows); renamed away by the
        // fully-unrolled loop.
        #pragma unroll
        for (int q = 0; q < 8; ++q) Bw[q] = Bw[q + 4];
    }
}

// ---------------------------------------------------------------------------
extern "C" void kernel_launch(void* const* d_in, const int* in_sizes, int n_in,
                              void* d_out, int out_size, void* d_ws, size_t ws_size,
                              hipStream_t stream) {
    const float* x      = (const float*)d_in[0];  // [T,B,C]
    const float* weight = (const float*)d_in[1];  // [H,1,K]
    const float* bias   = (const float*)d_in[2];  // [C]
    float*       out    = (float*)d_out;
    float*       sw     = (float*)d_ws;           // H*K softmaxed weights

    lwconv_softmax_w<<<1, 32, 0, stream>>>(weight, sw);
    lwconv_wmma_kernel<<<GRID_BLOCKS, 256, 0, stream>>>(x, sw, bias, out);
}